// RelationAwareMultiheadAttention_89232240542274
// MI455X (gfx1250) — compile-verified
//
#include <hip/hip_runtime.h>

// Problem constants (from reference setup_inputs)
#define B_   2
#define S_   512
#define E_   256
#define H_   8
#define D_   32
#define NREL 33

typedef __attribute__((ext_vector_type(2))) float v2f;
typedef __attribute__((ext_vector_type(8))) float v8f;

// ---------------------------------------------------------------------------
// Kernel 1: q = x @ W_Q^T and v = x @ W_V^T via V_WMMA_F32_16X16X4_F32.
// One wave computes one 16x16 output tile; K=256 in 64 steps of 4.
//
// A (16x4 f32) layout: lanes 0-15 -> M=lane, VGPR0=K0,VGPR1=K1;
//                      lanes 16-31 -> M=lane-16, VGPR0=K2,VGPR1=K3.
//   -> per-lane float2 load at X[(m0+l)*E + e + 2*half].
// B (4x16 f32) layout: lanes 0-15 -> N=lane, rows K0(K2 for hi half) in VGPR0;
//   B[k][n] = W[n][k] (x @ W^T) -> float2 load at W[(n0+l)*E + e + 2*half].
// C/D: VGPR r, lanes 0-15 -> (M=r, N=lane); lanes 16-31 -> (M=8+r, N=lane-16).
// ---------------------------------------------------------------------------
__global__ __launch_bounds__(256) void proj_wmma_f32(
    const float* __restrict__ query,
    const float* __restrict__ value,
    const float* __restrict__ Wq,
    const float* __restrict__ Wv,
    float* __restrict__ qout,
    float* __restrict__ vout)
{
    const int wave = (int)((blockIdx.x * blockDim.x + threadIdx.x) >> 5);
    const int lane = (int)(threadIdx.x & 31u);

    const int tilesPerProj = (B_ * S_ / 16) * (E_ / 16); // 64 * 16 = 1024
    const int p  = wave / tilesPerProj;                  // 0 = q-proj, 1 = v-proj
    const int r  = wave % tilesPerProj;
    const int mt = r / (E_ / 16);
    const int nt = r % (E_ / 16);

    const float* X   = (p == 0) ? query : value;
    const float* W   = (p == 0) ? Wq    : Wv;
    float*       OUT = (p == 0) ? qout  : vout;

    const int half = lane >> 4;   // 0: K-pair {0,1}; 1: K-pair {2,3}
    const int l    = lane & 15;

    const float* arow = X + (mt * 16 + l) * E_ + 2 * half;
    const float* brow = W + (nt * 16 + l) * E_ + 2 * half;

    v8f acc = {};
#pragma unroll 8
    for (int e = 0; e < E_; e += 4) {
        v2f a = *(const v2f*)(arow + e);
        v2f b = *(const v2f*)(brow + e);
        acc = __builtin_amdgcn_wmma_f32_16x16x4_f32(
            /*neg_a=*/false, a, /*neg_b=*/false, b,
            /*c_mod=*/(short)0, acc, /*reuse_a=*/false, /*reuse_b=*/false);
    }

#pragma unroll
    for (int rr = 0; rr < 8; ++rr) {
        OUT[(mt * 16 + rr + 8 * half) * E_ + nt * 16 + l] = acc[rr];
    }
}

// ---------------------------------------------------------------------------
// Kernel 2: per (b,i) row — histogram of relations, 33-bin scores/softmax,
// and output blend  z = v + sum_t w_t * rel_emb[t].
// One 256-thread block per row (grid = B*S = 1024).
// ---------------------------------------------------------------------------
__global__ __launch_bounds__(256) void rel_attn_fused(
    const float* __restrict__ q,        // [B*S, E] projected q
    const float* __restrict__ v,        // [B*S, E] projected v
    const int*   __restrict__ rel,      // [B*S, S] relation ids in [0,33)
    const float* __restrict__ rel_emb,  // [33, E]
    float* __restrict__ out)            // [B*S, E]
{
    const int row = (int)blockIdx.x;
    const int tid = (int)threadIdx.x;

    __shared__ float q_s[E_];
    __shared__ int   cnt[NREL];
    __shared__ float sw[H_ * NREL];     // scores, then normalized weights

    q_s[tid] = q[row * E_ + tid];
    const float v_val = v[row * E_ + tid];
    if (tid < NREL) cnt[tid] = 0;
    __syncthreads();

    // Histogram of this row's relation ids (LDS atomics, 512 entries).
    const int* rrow = rel + row * S_;
    for (int j = tid; j < S_; j += 256)
        atomicAdd(&cnt[rrow[j]], 1);
    __syncthreads();

    // Scores s[h][t] = scale * (q[h,:] . rel_emb[t, h*D:(h+1)*D])
    const float scale = 0.17677669529663687f;  // 1/sqrt(D=32)
    for (int idx = tid; idx < H_ * NREL; idx += 256) {
        const int h = idx / NREL, t = idx % NREL;
        const float* re = rel_emb + t * E_ + h * D_;
        const float* qs = q_s + h * D_;
        float dot = 0.f;
#pragma unroll
        for (int d = 0; d < D_; ++d) dot += qs[d] * re[d];
        sw[idx] = dot * scale;
    }
    __syncthreads();

    // Per-head 33-bin softmax weighted by bin counts:
    //   w_t = cnt_t * exp(s_t - m) / Z   (cnt_t == 0 contributes nothing)
    if (tid < H_) {
        float m = -3.4e38f;
        for (int t = 0; t < NREL; ++t) m = fmaxf(m, sw[tid * NREL + t]);
        float Z = 0.f;
        for (int t = 0; t < NREL; ++t)
            Z += (float)cnt[t] * __expf(sw[tid * NREL + t] - m);
        const float inv = 1.f / Z;
        for (int t = 0; t < NREL; ++t)
            sw[tid * NREL + t] = (float)cnt[t] * __expf(sw[tid * NREL + t] - m) * inv;
    }
    __syncthreads();

    // z[e] = v[e] + sum_t w[h][t] * rel_emb[t][e]   (h = e / D)
    const int h = tid / D_;
    const float* wv = sw + h * NREL;
    float acc = v_val;
#pragma unroll
    for (int t = 0; t < NREL; ++t)
        acc += wv[t] * rel_emb[t * E_ + tid];
    out[row * E_ + tid] = acc;
}

// ---------------------------------------------------------------------------
// Launch. Inputs (setup_inputs order):
//   0 query  1 key(unused: q.k_i term is j-independent -> cancels in softmax)
//   2 value  3 relations  4 W_Q  5 W_K(unused)  6 W_V  7 rel_emb
// Workspace: q buffer (1 MB) + v buffer (1 MB).
// ---------------------------------------------------------------------------
extern "C" void kernel_launch(void* const* d_in, const int* in_sizes, int n_in,
                              void* d_out, int out_size, void* d_ws, size_t ws_size,
                              hipStream_t stream)
{
    (void)in_sizes; (void)n_in; (void)out_size; (void)ws_size;

    const float* query = (const float*)d_in[0];
    const float* value = (const float*)d_in[2];
    const int*   rel   = (const int*)  d_in[3];
    const float* Wq    = (const float*)d_in[4];
    const float* Wv    = (const float*)d_in[6];
    const float* remb  = (const float*)d_in[7];

    float* qbuf = (float*)d_ws;
    float* vbuf = qbuf + (size_t)B_ * S_ * E_;

    const int tiles  = 2 * (B_ * S_ / 16) * (E_ / 16); // 2048 wave-tiles
    const int blocks = tiles / 8;                      // 8 waves (256 thr) per block

    proj_wmma_f32<<<blocks, 256, 0, stream>>>(query, value, Wq, Wv, qbuf, vbuf);
    rel_attn_fused<<<B_ * S_, 256, 0, stream>>>(qbuf, vbuf, rel, remb, (float*)d_out);
}